// RFCM_loss_37580963840096
// MI455X (gfx1250) — compile-verified
//
#include <hip/hip_runtime.h>
#include <cstdint>

// ---------------------------------------------------------------------------
// RFCM loss for MI455X (gfx1250).
// Bandwidth-bound (~170MB traffic / ~0.6 GFLOP) -> two-pass streaming design.
// CDNA5 path used: Tensor Data Mover async tile loads to LDS (TENSORcnt) to
// stage stencil tiles, overlapped with separable 3x3x3 box-filter compute.
// ---------------------------------------------------------------------------

#define NV (128 * 128 * 128)

// workspace layout (floats)
#define WS_V  0      // 8 centroids v[b*4+k]
#define WS_P1 8      // pass-1 partials: 8 channels * 64 blocks * 2
#define WS_P2 1032   // pass-2 partials: 256 blocks * 2

#define HT  8        // output rows (H) per tile
#define DCH 16       // output slices (D) per chunk

#if __has_builtin(__builtin_amdgcn_tensor_load_to_lds) && \
    __has_builtin(__builtin_amdgcn_s_wait_tensorcnt)
#define USE_TDM 1
#endif

typedef unsigned int v4u __attribute__((ext_vector_type(4)));
typedef int          v8i __attribute__((ext_vector_type(8)));
typedef int          v4i __attribute__((ext_vector_type(4)));

__device__ __forceinline__ unsigned lds_byte_offset(const void* p) {
  // generic -> LDS addrspace(3) cast; ptrtoint gives the 32-bit LDS byte offset
  return (unsigned)(uintptr_t)(__attribute__((address_space(3))) const void*)p;
}

#ifdef USE_TDM
// One 2D TDM descriptor: tile = nrows x 128 f32, row stride 128 elements,
// destination LDS byte offset lds_off. Packed per CDNA5 ISA section 8 (D#).
// This toolchain's builtin takes 6 args: (g0 v4u, g1 v8i, g2 v4i, g3 v4i,
// extra v8i group, i32 cpol) -- extra group zero-filled.
__device__ __forceinline__ void tdm_load_rows(const float* gsrc, unsigned lds_off,
                                              int nrows) {
  unsigned long long ga = (unsigned long long)(uintptr_t)gsrc;
  v4u g0;
  g0[0] = 1u;                                                 // count=1, user D#
  g0[1] = lds_off;                                            // lds_addr (bytes)
  g0[2] = (unsigned)ga;                                       // global_addr[31:0]
  g0[3] = (unsigned)((ga >> 32) & 0x01FFFFFFu) | (2u << 30);  // addr[56:32]|type=2
  v8i g1;
  g1[0] = (int)(2u << 16);               // workgroup_mask=0, data_size=2 (4B)
  g1[1] = (int)(128u << 16);             // tensor_dim0[15:0]=128
  g1[2] = (int)((unsigned)nrows << 16);  // tensor_dim0 hi=0 | tensor_dim1 lo=nrows
  g1[3] = (int)(128u << 16);             // tensor_dim1 hi=0 | tile_dim0=128
  g1[4] = nrows;                         // tile_dim1=nrows | tile_dim2=0 (2D)
  g1[5] = 128;                           // tensor_dim0_stride lo (elements)
  g1[6] = (int)(16384u << 16);           // stride0 hi=0 | stride1[15:0]=16384
  g1[7] = 0;                             // stride1[47:16]
  v4i g2; g2[0] = 1; g2[1] = 1; g2[2] = 0; g2[3] = 0;
  v4i g3; g3[0] = 0; g3[1] = 0; g3[2] = 0; g3[3] = 0;
  v8i g4; g4[0] = 0; g4[1] = 0; g4[2] = 0; g4[3] = 0;
          g4[4] = 0; g4[5] = 0; g4[6] = 0; g4[7] = 0;
  __builtin_amdgcn_tensor_load_to_lds(g0, g1, g2, g3, g4, 0);
}
#endif

// deterministic 256-thread (8 wave32) block reduction; result valid in thread 0
__device__ __forceinline__ float blockReduceSum(float v, float* sbuf) {
#pragma unroll
  for (int o = 16; o > 0; o >>= 1) v += __shfl_down(v, o, 32);
  const int lane = (int)threadIdx.x & 31;
  const int wid  = (int)threadIdx.x >> 5;
  __syncthreads();
  if (lane == 0) sbuf[wid] = v;
  __syncthreads();
  if (wid == 0) {
    v = (lane < 8) ? sbuf[lane] : 0.f;
#pragma unroll
    for (int o = 4; o > 0; o >>= 1) v += __shfl_down(v, o, 32);
  }
  return v;
}

// ---------------------------------------------------------------------------
// Pass 1: per-(b,k) partial sums of mem and img*mem (for centroids)
// grid (64 blocks over N, 8 channels), block 256
// ---------------------------------------------------------------------------
__global__ __launch_bounds__(256) void k1_partials(const float* __restrict__ pred,
                                                   const float* __restrict__ img,
                                                   float* __restrict__ ws) {
  const int ch  = blockIdx.y;   // b*4 + k
  const int blk = blockIdx.x;
  const float4* p4 = (const float4*)(pred + (size_t)ch * NV);
  const float4* i4 = (const float4*)(img + (size_t)(ch >> 2) * NV);
  const int chunk = (NV / 4) / 64;  // 8192 float4 per block
  const int base  = blk * chunk;
  float sM = 0.f, sI = 0.f;
  for (int i = base + (int)threadIdx.x; i < base + chunk; i += 256) {
    float4 p = p4[i];
    float4 m = i4[i];
    float m0 = p.x * p.x, m1 = p.y * p.y, m2 = p.z * p.z, m3 = p.w * p.w;
    sM += (m0 + m1) + (m2 + m3);
    sI += (m0 * m.x + m1 * m.y) + (m2 * m.z + m3 * m.w);
  }
  __shared__ float sred[8];
  sM = blockReduceSum(sM, sred);
  sI = blockReduceSum(sI, sred);
  if (threadIdx.x == 0) {
    ws[WS_P1 + (ch * 64 + blk) * 2 + 0] = sM;
    ws[WS_P1 + (ch * 64 + blk) * 2 + 1] = sI;
  }
}

// deterministic centroid finalize: v[b,k] = sum(img*mem)/sum(mem)
__global__ void k1b_centroids(float* __restrict__ ws) {
  const int k = (int)threadIdx.x;
  if (k < 8) {
    float sM = 0.f, sI = 0.f;
    for (int i = 0; i < 64; ++i) {
      sM += ws[WS_P1 + (k * 64 + i) * 2 + 0];
      sI += ws[WS_P1 + (k * 64 + i) * 2 + 1];
    }
    ws[WS_V + k] = sI / sM;
  }
}

// ---------------------------------------------------------------------------
// Pass 2: per-voxel J1/J2 with separable 3x3x3 box filter.
// WG tile: HT rows x 128 cols x DCH slices. TDM stages (HT+2)x128 y_pred rows
// per cluster per slice into double-buffered LDS; D-sum kept in registers.
// grid (128/HT, 128/DCH, 2), block 256
// ---------------------------------------------------------------------------
__global__ __launch_bounds__(256) void k2_loss(const float* __restrict__ pred,
                                               const float* __restrict__ img,
                                               float* __restrict__ ws) {
  __shared__ float stage[2][4][(HT + 2) * 128];  // 40 KB
  __shared__ float sred[8];

  const int h0  = blockIdx.x * HT;
  const int d0  = blockIdx.y * DCH;
  const int b   = blockIdx.z;
  const int tid = (int)threadIdx.x;
  const int w   = tid & 127;   // fixed column per thread (coalesced / bank-free)
  const int t   = tid >> 7;    // row parity; thread owns rows t, t+2, t+4, t+6

  // clipped staging row range [rowStart, rowEnd) of the H-halo'ed tile
  const int rowStart = (h0 == 0) ? 0 : (h0 - 1);
  const int rowEnd   = (h0 + HT + 1 <= 128) ? (h0 + HT + 1) : 128;
  const int nrows    = rowEnd - rowStart;
  const int ldsRow   = (h0 == 0) ? 1 : 0;

  // zero halo rows the loader never writes (zero padding of the conv)
  if (tid < 128) {
    for (int c = 0; c < 8; ++c) {
      float* s = &stage[c >> 2][c & 3][0];
      if (h0 == 0) s[tid] = 0.f;
      if (h0 + HT == 128) s[(HT + 1) * 128 + tid] = 0.f;
    }
  }

  const int dFirst = d0 - 1, dLast = d0 + DCH;

  // preload slice dFirst (if it exists)
  if (dFirst >= 0) {
#ifdef USE_TDM
    if (__builtin_amdgcn_readfirstlane((int)threadIdx.x) == 0) {  // wave 0 only
      for (int k = 0; k < 4; ++k)
        tdm_load_rows(
            pred + (((size_t)(b * 4 + k) * 128 + dFirst) * 128 + rowStart) * 128,
            lds_byte_offset(&stage[dFirst & 1][k][ldsRow * 128]), nrows);
      __builtin_amdgcn_s_wait_tensorcnt(0);
    }
#else
    for (int k = 0; k < 4; ++k) {
      const float* src =
          pred + (((size_t)(b * 4 + k) * 128 + dFirst) * 128 + rowStart) * 128;
      float* dst = &stage[dFirst & 1][k][ldsRow * 128];
      for (int i = tid; i < nrows * 128; i += 256) dst[i] = src[i];
    }
#endif
  }
  __syncthreads();

  float vk[4];
  for (int k = 0; k < 4; ++k) vk[k] = ws[WS_V + b * 4 + k];

  // register rings: B* = HxW box-summed mem per slice; memp = mem at slice d-1
  float B0[4][4], B1[4][4], memp[4][4];
  for (int k = 0; k < 4; ++k)
    for (int j = 0; j < 4; ++j) { B0[k][j] = 0.f; B1[k][j] = 0.f; memp[k][j] = 0.f; }
  float accJ1 = 0.f, accJ2 = 0.f;

  for (int d = dFirst; d <= dLast; ++d) {
    const int dn = d + 1;
    const bool havenext = (dn <= dLast) && (dn < 128);

#ifdef USE_TDM
    // async prefetch of next slice overlaps this iteration's compute
    if (havenext) {
      if (__builtin_amdgcn_readfirstlane((int)threadIdx.x) == 0) {
        for (int k = 0; k < 4; ++k)
          tdm_load_rows(
              pred + (((size_t)(b * 4 + k) * 128 + dn) * 128 + rowStart) * 128,
              lds_byte_offset(&stage[dn & 1][k][ldsRow * 128]), nrows);
      }
    }
#endif

    // B[d] = H-sum3(W-sum3(mem[d])) at this thread's 4 voxels; memc = mem[d]
    float B2[4][4], memc[4][4];
    if (d >= 0 && d < 128) {
      const int buf = d & 1;
      for (int k = 0; k < 4; ++k) {
        float wsum[9], m2c[9];
#pragma unroll
        for (int r = 0; r < 9; ++r) {
          const float* row = &stage[buf][k][(t + r) * 128];
          float bm = row[w];
          float b2 = bm * bm;
          float a2 = 0.f, c2 = 0.f;
          if (w > 0)   { float a = row[w - 1]; a2 = a * a; }
          if (w < 127) { float c = row[w + 1]; c2 = c * c; }
          wsum[r] = (a2 + c2) + b2;
          m2c[r]  = b2;
        }
#pragma unroll
        for (int j = 0; j < 4; ++j) {
          B2[k][j]   = wsum[2 * j] + wsum[2 * j + 1] + wsum[2 * j + 2];
          memc[k][j] = m2c[2 * j + 1];
        }
      }
    } else {
      for (int k = 0; k < 4; ++k)
        for (int j = 0; j < 4; ++j) { B2[k][j] = 0.f; memc[k][j] = 0.f; }
    }

    // emit output voxels at depth dc = d-1 (box27 = B[dc-1]+B[dc]+B[dc+1])
    const int dc = d - 1;
    if (dc >= d0) {
#pragma unroll
      for (int j = 0; j < 4; ++j) {
        const int h = h0 + t + 2 * j;
        const float iv = img[(((size_t)b * 128 + dc) * 128 + h) * 128 + w];
        float nb[4], Snb = 0.f;
#pragma unroll
        for (int k = 0; k < 4; ++k) {
          nb[k] = ((B0[k][j] + B1[k][j]) + B2[k][j]) - memp[k][j];
          Snb += nb[k];
        }
#pragma unroll
        for (int k = 0; k < 4; ++k) {
          const float m  = memp[k][j];
          const float dv = iv - vk[k];
          accJ1 += m * (dv * dv);
          accJ2 += m * (Snb - nb[k]);
        }
      }
    }

    // rotate rings
    for (int k = 0; k < 4; ++k)
      for (int j = 0; j < 4; ++j) {
        B0[k][j]   = B1[k][j];
        B1[k][j]   = B2[k][j];
        memp[k][j] = memc[k][j];
      }

#ifdef USE_TDM
    if (havenext) {
      if (__builtin_amdgcn_readfirstlane((int)threadIdx.x) == 0)
        __builtin_amdgcn_s_wait_tensorcnt(0);
    }
#else
    if (havenext) {
      for (int k = 0; k < 4; ++k) {
        const float* src =
            pred + (((size_t)(b * 4 + k) * 128 + dn) * 128 + rowStart) * 128;
        float* dst = &stage[dn & 1][k][ldsRow * 128];
        for (int i = tid; i < nrows * 128; i += 256) dst[i] = src[i];
      }
    }
#endif
    __syncthreads();
  }

  accJ1 = blockReduceSum(accJ1, sred);
  accJ2 = blockReduceSum(accJ2, sred);
  if (threadIdx.x == 0) {
    const int blin =
        ((int)blockIdx.z * (int)gridDim.y + (int)blockIdx.y) * (int)gridDim.x +
        (int)blockIdx.x;
    ws[WS_P2 + blin * 2 + 0] = accJ1;
    ws[WS_P2 + blin * 2 + 1] = accJ2;
  }
}

// ---------------------------------------------------------------------------
// Final deterministic tree reduction of 256 block partials -> scalar loss
// ---------------------------------------------------------------------------
__global__ __launch_bounds__(256) void k3_final(const float* __restrict__ ws,
                                                float* __restrict__ out) {
  __shared__ float s1[256], s2[256];
  const int tc = (int)threadIdx.x;
  s1[tc] = ws[WS_P2 + tc * 2 + 0];
  s2[tc] = ws[WS_P2 + tc * 2 + 1];
  __syncthreads();
  for (int o = 128; o > 0; o >>= 1) {
    if (tc < o) { s1[tc] += s1[tc + o]; s2[tc] += s2[tc + o]; }
    __syncthreads();
  }
  if (tc == 0) {
    const float inv = 1.0f / (float)(2 * NV);  // mean over B*N
    out[0] = s1[0] * inv + 0.0008f * (s2[0] * inv);
  }
}

extern "C" void kernel_launch(void* const* d_in, const int* in_sizes, int n_in,
                              void* d_out, int out_size, void* d_ws, size_t ws_size,
                              hipStream_t stream) {
  (void)in_sizes; (void)n_in; (void)out_size; (void)ws_size;
  const float* pred = (const float*)d_in[0];  // y_pred [2,4,128,128,128] f32
  const float* img  = (const float*)d_in[1];  // image  [2,1,128,128,128] f32
  float* out = (float*)d_out;                 // scalar f32
  float* ws  = (float*)d_ws;                  // >= 1544 floats used

  k1_partials<<<dim3(64, 8, 1), 256, 0, stream>>>(pred, img, ws);
  k1b_centroids<<<dim3(1, 1, 1), 32, 0, stream>>>(ws);
  k2_loss<<<dim3(128 / HT, 128 / DCH, 2), 256, 0, stream>>>(pred, img, ws);
  k3_final<<<dim3(1, 1, 1), 256, 0, stream>>>(ws, out);
}